// Encoder_30743375904797
// MI455X (gfx1250) — compile-verified
//
#include <hip/hip_runtime.h>
#include <hip/hip_bf16.h>

// ---------------------------------------------------------------------------
// VGAE / GCN encoder for MI455X (gfx1250, wave32).
//   hidden  = softplus( Agg( x @ W1 ) + b1 )
//   z_loc   =           Agg( hidden @ W_mu  ) + b_mu
//   z_scale =     exp(  Agg( hidden @ W_sig ) + b_sig )
// Agg(v)[d] = sum_{(s,d) in E+selfloops} v[s] * dinv[s]*dinv[d]
// GEMMs use the native fp32 matrix path V_WMMA_F32_16X16X4_F32.
// ---------------------------------------------------------------------------

typedef __attribute__((ext_vector_type(2))) float v2f;
typedef __attribute__((ext_vector_type(8))) float v8f;

// ---------------- WMMA fp32 GEMM: C[M,N] = A[M,K] @ B[K,N] ------------------
// One wave computes a 16-row stripe across the full N width (NTILES 16-col
// tiles), so the A fragment is loaded once per K-step and reused for all
// column tiles. Layouts per CDNA5 ISA 7.12.2:
//   A 16x4 : lane L holds A[M = L%16][K = 2*(L/16) + {0,1}] in v[0],v[1]
//   B 4x16 : lane L holds B[K = 2*(L/16) + {0,1}][N = L%16] in v[0],v[1]
//   C 16x16: lane L, VGPR v holds C[M = v + 8*(L/16)][N = L%16]
// Bounds check is wave-uniform so EXEC is all-1s at every WMMA.
template <int K, int NTILES>
__global__ __launch_bounds__(256) void wmma_gemm_f32(
    const float* __restrict__ A, const float* __restrict__ B,
    float* __restrict__ C, int Mtiles) {
  constexpr int N = NTILES * 16;
  const int lane  = threadIdx.x & 31;
  const int wave  = threadIdx.x >> 5;
  const int mtile = blockIdx.x * (blockDim.x >> 5) + wave;
  if (mtile >= Mtiles) return;  // uniform per wave -> EXEC stays all-1s

  const int ln   = lane & 15;
  const int half = lane >> 4;  // 0: K pair {0,1}, 1: K pair {2,3}
  const size_t arow = (size_t)(mtile * 16 + ln) * (size_t)K;

  v8f acc[NTILES];
#pragma unroll
  for (int t = 0; t < NTILES; ++t)
    acc[t] = (v8f){0.f, 0.f, 0.f, 0.f, 0.f, 0.f, 0.f, 0.f};

  for (int k = 0; k < K; k += 4) {
    const int kb = k + half * 2;
    v2f a;
    a.x = A[arow + kb];
    a.y = A[arow + kb + 1];
#pragma unroll
    for (int t = 0; t < NTILES; ++t) {
      v2f b;
      b.x = B[(size_t)kb * N + t * 16 + ln];
      b.y = B[(size_t)(kb + 1) * N + t * 16 + ln];
      acc[t] = __builtin_amdgcn_wmma_f32_16x16x4_f32(
          /*neg_a=*/false, a, /*neg_b=*/false, b,
          /*c_mod=*/(short)0, acc[t], /*reuse_a=*/false, /*reuse_b=*/false);
    }
  }

  const size_t mrow = (size_t)(mtile * 16 + 8 * half);
#pragma unroll
  for (int t = 0; t < NTILES; ++t)
#pragma unroll
    for (int v = 0; v < 8; ++v)
      C[(mrow + v) * N + t * 16 + ln] = acc[t][v];
}

// ---------------- degree / normalization ------------------------------------
__global__ void fill_kernel(float* __restrict__ p, int n, float v) {
  int i = blockIdx.x * blockDim.x + threadIdx.x;
  if (i < n) p[i] = v;
}

__global__ void deg_accum_kernel(const int* __restrict__ dst,
                                 float* __restrict__ deg, int nE) {
  int i = blockIdx.x * blockDim.x + threadIdx.x;
  if (i < nE)
    __hip_atomic_fetch_add(&deg[dst[i]], 1.0f, __ATOMIC_RELAXED,
                           __HIP_MEMORY_SCOPE_AGENT);
}

__global__ void rsqrt_kernel(float* __restrict__ p, int n) {
  int i = blockIdx.x * blockDim.x + threadIdx.x;
  if (i < n) p[i] = rsqrtf(p[i]);  // deg >= 1 always (self-loop)
}

// ---------------- aggregation ------------------------------------------------
// Self-loop contribution (also initializes the accumulator buffer):
template <int F>
__global__ void init_self_kernel(const float* __restrict__ feat,
                                 const float* __restrict__ dinv,
                                 float* __restrict__ out, size_t total) {
  size_t i = (size_t)blockIdx.x * blockDim.x + threadIdx.x;
  if (i >= total) return;
  float di = dinv[i / F];
  out[i] = feat[i] * di * di;
}

// Edge scatter-add: out[dst] += feat[src] * dinv[src]*dinv[dst]
template <int F>
__global__ __launch_bounds__(256) void scatter_edges_kernel(
    const int* __restrict__ src, const int* __restrict__ dst,
    const float* __restrict__ dinv, const float* __restrict__ feat,
    float* __restrict__ out, int nE) {
  constexpr int EPB = 256 / F;  // edges per 256-thread block
  const int e = blockIdx.x * EPB + (threadIdx.x / F);
  const int f = threadIdx.x % F;
  if (e >= nE) return;
  const int s = src[e];
  const int d = dst[e];
  const float nm = dinv[s] * dinv[d];
  const float v = feat[(size_t)s * F + f] * nm;
  __hip_atomic_fetch_add(&out[(size_t)d * F + f], v, __ATOMIC_RELAXED,
                         __HIP_MEMORY_SCOPE_AGENT);
}

// ---------------- pointwise epilogues ---------------------------------------
template <int F>
__global__ void softplus_bias_kernel(float* __restrict__ io,
                                     const float* __restrict__ b,
                                     size_t total) {
  size_t i = (size_t)blockIdx.x * blockDim.x + threadIdx.x;
  if (i >= total) return;
  float v = io[i] + b[i % F];
  // numerically stable softplus = max(v,0) + log1p(exp(-|v|))
  io[i] = fmaxf(v, 0.0f) + log1pf(expf(-fabsf(v)));
}

template <int F>
__global__ void add_bias_kernel(float* __restrict__ io,
                                const float* __restrict__ b, size_t total) {
  size_t i = (size_t)blockIdx.x * blockDim.x + threadIdx.x;
  if (i >= total) return;
  io[i] += b[i % F];
}

template <int F>
__global__ void exp_bias_kernel(float* __restrict__ io,
                                const float* __restrict__ b, size_t total) {
  size_t i = (size_t)blockIdx.x * blockDim.x + threadIdx.x;
  if (i >= total) return;
  io[i] = expf(io[i] + b[i % F]);
}

// ---------------------------------------------------------------------------
extern "C" void kernel_launch(void* const* d_in, const int* in_sizes, int n_in,
                              void* d_out, int out_size, void* d_ws,
                              size_t ws_size, hipStream_t stream) {
  const float* x    = (const float*)d_in[0];
  const int*   eidx = (const int*)d_in[1];   // [2, E] (src row, dst row)
  const float* W1   = (const float*)d_in[2]; // [256,128]
  const float* b1   = (const float*)d_in[3];
  const float* Wmu  = (const float*)d_in[4]; // [128,64]
  const float* bmu  = (const float*)d_in[5];
  const float* Wsig = (const float*)d_in[6]; // [128,64]
  const float* bsig = (const float*)d_in[7];

  const int N = in_sizes[0] / 256;  // 100000 nodes
  const int E = in_sizes[1] / 2;    // 1600000 edges
  const int* src = eidx;
  const int* dst = eidx + E;

  // Workspace layout (bytes):
  //   [0, 4N)            dinv (degree, then rsqrt in place)
  //   [1MB, 1MB+512N)    buf1: h (N x 128), later t_mu / t_sig (N x 64)
  //   [1MB+512N, +512N)  buf2: agg1 / hidden (N x 128)
  char*  ws   = (char*)d_ws;
  float* dinv = (float*)ws;
  float* buf1 = (float*)(ws + (size_t)(1u << 20));
  float* buf2 = (float*)(ws + (size_t)(1u << 20) + (size_t)N * 128 * 4);
  float* zloc = (float*)d_out;
  float* zsig = zloc + (size_t)N * 64;

  const dim3 blk(256);
  const int gN  = (N + 255) / 256;
  const int gE  = (E + 255) / 256;
  const int Mtiles = N / 16;                 // 6250
  const int gG  = (Mtiles + 7) / 8;          // 8 waves per 256-thread block
  const size_t tot1 = (size_t)N * 128;
  const size_t tot2 = (size_t)N * 64;
  const int g1 = (int)((tot1 + 255) / 256);
  const int g2 = (int)((tot2 + 255) / 256);

  // --- GCN normalization: deg (with self-loops) -> dinv -------------------
  fill_kernel<<<gN, blk, 0, stream>>>(dinv, N, 1.0f);
  deg_accum_kernel<<<gE, blk, 0, stream>>>(dst, dinv, E);
  rsqrt_kernel<<<gN, blk, 0, stream>>>(dinv, N);

  // --- layer 1: h = x @ W1 ; agg ; softplus -------------------------------
  wmma_gemm_f32<256, 8><<<gG, blk, 0, stream>>>(x, W1, buf1, Mtiles);
  init_self_kernel<128><<<g1, blk, 0, stream>>>(buf1, dinv, buf2, tot1);
  scatter_edges_kernel<128><<<(E + 1) / 2, blk, 0, stream>>>(src, dst, dinv,
                                                             buf1, buf2, E);
  softplus_bias_kernel<128><<<g1, blk, 0, stream>>>(buf2, b1, tot1);

  // --- z_loc = Agg(hidden @ W_mu) + b_mu ----------------------------------
  wmma_gemm_f32<128, 4><<<gG, blk, 0, stream>>>(buf2, Wmu, buf1, Mtiles);
  init_self_kernel<64><<<g2, blk, 0, stream>>>(buf1, dinv, zloc, tot2);
  scatter_edges_kernel<64><<<(E + 3) / 4, blk, 0, stream>>>(src, dst, dinv,
                                                            buf1, zloc, E);
  add_bias_kernel<64><<<g2, blk, 0, stream>>>(zloc, bmu, tot2);

  // --- z_scale = exp(Agg(hidden @ W_sig) + b_sig) -------------------------
  wmma_gemm_f32<128, 4><<<gG, blk, 0, stream>>>(buf2, Wsig, buf1, Mtiles);
  init_self_kernel<64><<<g2, blk, 0, stream>>>(buf1, dinv, zsig, tot2);
  scatter_edges_kernel<64><<<(E + 3) / 4, blk, 0, stream>>>(src, dst, dinv,
                                                            buf1, zsig, E);
  exp_bias_kernel<64><<<g2, blk, 0, stream>>>(zsig, bsig, tot2);
}